// Generator_4569845203153
// MI455X (gfx1250) — compile-verified
//
#include <hip/hip_runtime.h>

typedef __attribute__((ext_vector_type(2))) float v2f;
typedef __attribute__((ext_vector_type(8))) float v8f;

#define T_STEPS 65536
#define ZDIM 100
#define HDIM 100
#define G4 400   // 4*H

// ---------------------------------------------------------------------------
// Stage A: y = relu(relu(inputs @ W_z + b_z) @ W_y + b_y)
// One wave32 per row t. Coalesced row loads, butterfly reduction for the
// 10-wide z vector, then each lane produces y[j] for j = lane, lane+32, ...
// ---------------------------------------------------------------------------
__global__ __launch_bounds__(256) void dense_front(
    const float* __restrict__ in, const float* __restrict__ Wz,
    const float* __restrict__ bz, const float* __restrict__ Wy,
    const float* __restrict__ by, float* __restrict__ y) {
  const int lane = threadIdx.x & 31;
  const int row  = (blockIdx.x * blockDim.x + threadIdx.x) >> 5;  // one wave = one t
  if (row >= T_STEPS) return;
  const float* x = in + (size_t)row * ZDIM;

  float p[10];
#pragma unroll
  for (int j = 0; j < 10; ++j) p[j] = 0.f;
  for (int k = lane; k < ZDIM; k += 32) {
    const float xv = x[k];
#pragma unroll
    for (int j = 0; j < 10; ++j) p[j] = fmaf(xv, Wz[k * 10 + j], p[j]);
  }
#pragma unroll
  for (int j = 0; j < 10; ++j) {
    float v = p[j];
#pragma unroll
    for (int s = 16; s; s >>= 1) v += __shfl_xor(v, s, 32);
    p[j] = fmaxf(v + bz[j], 0.f);                 // z (all lanes hold full vector)
  }
  for (int j = lane; j < HDIM; j += 32) {
    float acc = by[j];
#pragma unroll
    for (int q = 0; q < 10; ++q) acc = fmaf(p[q], Wy[q * HDIM + j], acc);
    y[(size_t)row * HDIM + j] = fmaxf(acc, 0.f);
  }
}

// ---------------------------------------------------------------------------
// Stage B: xg = y @ lstm_kernel + lstm_bias  via V_WMMA_F32_16X16X4_F32.
// One wave per 16x16 output tile, K = 100 -> 25 WMMA K-steps of 4.
// A (16x4 f32): lanes 0-15 carry K={0,1}, lanes 16-31 carry K={2,3}.
// C/D (16x16 f32): VGPR v -> M = v (lanes 0-15) / v+8 (lanes 16-31).
// ---------------------------------------------------------------------------
__global__ __launch_bounds__(256) void gemm_xg(
    const float* __restrict__ y, const float* __restrict__ K,
    const float* __restrict__ bias, float* __restrict__ xg) {
  const int lane = threadIdx.x & 31;
  const int tile = (blockIdx.x * blockDim.x + threadIdx.x) >> 5;
  const int NT = G4 / 16;                                  // 25 N-tiles
  if (tile >= (T_STEPS / 16) * NT) return;
  const int mb   = (tile / NT) * 16;
  const int nb   = (tile % NT) * 16;
  const int half = lane >> 4;                              // 0: lanes 0-15, 1: 16-31
  const int l16  = lane & 15;

  v8f c = {};
  for (int k0 = 0; k0 < HDIM; k0 += 4) {
    const int ka = k0 + (half ? 2 : 0);
    v2f a, b;
    a.x = y[(size_t)(mb + l16) * HDIM + ka];
    a.y = y[(size_t)(mb + l16) * HDIM + ka + 1];
    b.x = K[(size_t)(ka)     * G4 + nb + l16];
    b.y = K[(size_t)(ka + 1) * G4 + nb + l16];
    c = __builtin_amdgcn_wmma_f32_16x16x4_f32(
        /*neg_a=*/false, a, /*neg_b=*/false, b,
        /*c_mod=*/(short)0, c, /*reuse_a=*/false, /*reuse_b=*/false);
  }
  const float bn = bias[nb + l16];
#pragma unroll
  for (int v = 0; v < 8; ++v) {
    const int m = mb + v + (half ? 8 : 0);
    xg[(size_t)m * G4 + nb + l16] = c[v] + bn;
  }
}

// ---------------------------------------------------------------------------
// Stage C: serial LSTM scan. Single workgroup, 416 threads (13 wave32s).
// Thread j < 400 owns column j of lstm_recurrent in registers (100 VGPRs).
// h lives in LDS (broadcast reads); gates exchanged through LDS; xg row for
// t+1 is prefetched into a register one full step ahead of its use.
// ---------------------------------------------------------------------------
__global__ __launch_bounds__(416, 1) void lstm_scan(
    const float* __restrict__ xg, const float* __restrict__ R,
    float* __restrict__ hs) {
  const int tid = threadIdx.x;
  __shared__ __align__(16) float hbuf[104];
  __shared__ float gbuf[G4];

  float rc[HDIM];
  if (tid < G4) {
#pragma unroll
    for (int k = 0; k < HDIM; ++k) rc[k] = R[(size_t)k * G4 + tid];  // coalesced
  }
  if (tid < 104) hbuf[tid] = 0.f;
  float cst   = 0.f;
  float gnext = (tid < G4) ? xg[tid] : 0.f;   // prefetch t = 0
  __syncthreads();

  for (int t = 0; t < T_STEPS; ++t) {
    const float gcur = gnext;
    if (tid < G4 && t + 1 < T_STEPS)
      gnext = xg[(size_t)(t + 1) * G4 + tid];  // hide global latency behind step

    if (tid < G4) {
      float a0 = 0.f, a1 = 0.f, a2 = 0.f, a3 = 0.f;  // break dependent FMA chain
#pragma unroll
      for (int k = 0; k < HDIM; k += 4) {
        const float4 hv = *reinterpret_cast<const float4*>(&hbuf[k]);
        a0 = fmaf(hv.x, rc[k],     a0);
        a1 = fmaf(hv.y, rc[k + 1], a1);
        a2 = fmaf(hv.z, rc[k + 2], a2);
        a3 = fmaf(hv.w, rc[k + 3], a3);
      }
      const float g = gcur + ((a0 + a1) + (a2 + a3));
      float act;
      if (tid < 200 || tid >= 300) act = 1.f / (1.f + __expf(-g));  // i, f, o
      else                         act = fmaxf(g, 0.f);             // relu(cand)
      gbuf[tid] = act;
    }
    __syncthreads();

    if (tid < HDIM) {
      const float i  = gbuf[tid];
      const float f  = gbuf[tid + 100];
      const float cd = gbuf[tid + 200];
      const float o  = gbuf[tid + 300];
      cst = fmaf(f, cst, i * cd);            // c = f*c + i*relu(cand)
      const float h = o * fmaxf(cst, 0.f);   // h = o*relu(c)
      hbuf[tid] = h;
      hs[(size_t)t * HDIM + tid] = h;        // fire-and-forget store
    }
    __syncthreads();
  }
}

// ---------------------------------------------------------------------------
// Stage D: out[t] = hs[t] . W_f + b_f, one wave per row, butterfly reduce.
// ---------------------------------------------------------------------------
__global__ __launch_bounds__(256) void proj_out(
    const float* __restrict__ hs, const float* __restrict__ Wf,
    const float* __restrict__ bf, float* __restrict__ out) {
  const int lane = threadIdx.x & 31;
  const int t    = (blockIdx.x * blockDim.x + threadIdx.x) >> 5;
  if (t >= T_STEPS) return;
  float p = 0.f;
  for (int k = lane; k < HDIM; k += 32)
    p = fmaf(hs[(size_t)t * HDIM + k], Wf[k], p);
#pragma unroll
  for (int s = 16; s; s >>= 1) p += __shfl_xor(p, s, 32);
  if (lane == 0) out[t] = p + bf[0];
}

// ---------------------------------------------------------------------------
extern "C" void kernel_launch(void* const* d_in, const int* in_sizes, int n_in,
                              void* d_out, int out_size, void* d_ws, size_t ws_size,
                              hipStream_t stream) {
  (void)in_sizes; (void)n_in; (void)out_size; (void)ws_size;
  const float* inputs = (const float*)d_in[0];
  const float* Wz     = (const float*)d_in[1];
  const float* bz     = (const float*)d_in[2];
  const float* Wy     = (const float*)d_in[3];
  const float* by     = (const float*)d_in[4];
  const float* lk     = (const float*)d_in[5];
  const float* lr     = (const float*)d_in[6];
  const float* lb     = (const float*)d_in[7];
  const float* Wf     = (const float*)d_in[8];
  const float* bf     = (const float*)d_in[9];

  float* ws = (float*)d_ws;
  float* y  = ws;                                  // T*H floats
  float* xg = ws + (size_t)T_STEPS * HDIM;         // T*4H floats
  float* hs = y;                                   // reuse y's region after GEMM

  dense_front<<<T_STEPS / 8, 256, 0, stream>>>(inputs, Wz, bz, Wy, by, y);

  const int tiles = (T_STEPS / 16) * (G4 / 16);    // 102400 wave-tiles
  gemm_xg<<<(tiles + 7) / 8, 256, 0, stream>>>(y, lk, lb, xg);

  lstm_scan<<<1, 416, 0, stream>>>(xg, lr, hs);

  proj_out<<<T_STEPS / 8, 256, 0, stream>>>(hs, Wf, bf, (float*)d_out);
}